// RateDistAttenUser_4587025072174
// MI455X (gfx1250) — compile-verified
//
#include <hip/hip_runtime.h>
#include <math.h>

// Problem constants (match reference)
#define BB     512
#define TT     256
#define HH     256
#define DU     64
#define NC     64
#define KTOT   320                 // H + DIM_USR
#define ROWS   (BB * TT)           // 131072 rows of the fused GEMM
#define KCH    80                  // K chunk staged in LDS (transposed W)
#define NKCH   (KTOT / KCH)        // 4 chunks
#define TSTRIDE 68                 // padded LDS stride for 16x64 logit tile
#define WAVES  8                   // waves per block in the GEMM kernel
#define EPSF   1e-8f
#define LOG_EPS (-18.420680743952367f)   // logf(1e-8)

typedef __attribute__((ext_vector_type(2))) float v2f;
typedef __attribute__((ext_vector_type(8))) float v8f;

// -----------------------------------------------------------------------------
// Kernel 1: fused  logits = [x | x_user] @ W + b  -> softmax -> KL -> neg_kl
// One wave per 16-row tile, V_WMMA_F32_16X16X4_F32, K streamed in LDS chunks.
// -----------------------------------------------------------------------------
__global__ __launch_bounds__(256)
void rda_gemm_kl(const float* __restrict__ x,     // [ROWS][HH]
                 const float* __restrict__ xu,    // [ROWS][DU]
                 const float* __restrict__ dist,  // [ROWS][NC]
                 const float* __restrict__ W,     // [KTOT][NC]
                 const float* __restrict__ bias,  // [NC]
                 float* __restrict__ neg_kl)      // [ROWS]
{
    __shared__ float sW[NC * KCH];                 // W chunk, transposed: sW[n*KCH + k]
    __shared__ float sTile[WAVES * 16 * TSTRIDE];  // per-wave 16x64 logits (padded)

    const int tid  = threadIdx.x;
    const int lane = tid & 31;
    const int wv   = tid >> 5;
    const int lo   = lane & 15;   // M (A) / N (B,C)
    const int hf   = lane >> 4;   // K-half selector
    const int tile = blockIdx.x * WAVES + wv;
    const int row0 = tile * 16;
    const int arow = row0 + lo;

    v8f a0 = {}, a1 = {}, a2 = {}, a3 = {};

    for (int kc = 0; kc < NKCH; ++kc) {
        __syncthreads();  // previous chunk readers done before overwriting sW
        // cooperative transpose-load of W chunk: global coalesced, LDS [n][k]
        for (int i = 0; i < (KCH * NC) / 256; ++i) {
            int idx = i * 256 + tid;        // 0..5119 over chunk
            int kl  = idx >> 6;             // k within chunk
            int n   = idx & 63;
            sW[n * KCH + kl] = W[kc * (KCH * NC) + idx];
        }
        __syncthreads();

        const int kbase = kc * KCH;
        for (int kb = 0; kb < KCH; kb += 4) {
            const int kg = kbase + kb + 2 * hf;      // even -> 8B aligned pair
            v2f a;
            if (kg < HH) {
                a = *(const v2f*)(x + (long)arow * HH + kg);
            } else {
                a = *(const v2f*)(xu + (long)arow * DU + (kg - HH));
            }
            const int kl = kb + 2 * hf;
            v2f b0 = *(const v2f*)(&sW[(0 * 16 + lo) * KCH + kl]);
            v2f b1 = *(const v2f*)(&sW[(1 * 16 + lo) * KCH + kl]);
            v2f b2 = *(const v2f*)(&sW[(2 * 16 + lo) * KCH + kl]);
            v2f b3 = *(const v2f*)(&sW[(3 * 16 + lo) * KCH + kl]);
            a0 = __builtin_amdgcn_wmma_f32_16x16x4_f32(false, a, false, b0, (short)0, a0, false, false);
            a1 = __builtin_amdgcn_wmma_f32_16x16x4_f32(false, a, false, b1, (short)0, a1, false, false);
            a2 = __builtin_amdgcn_wmma_f32_16x16x4_f32(false, a, false, b2, (short)0, a2, false, false);
            a3 = __builtin_amdgcn_wmma_f32_16x16x4_f32(false, a, false, b3, (short)0, a3, false, false);
        }
    }

    // bias add + spill tile to LDS:  lane holds column N = t*16+lo, rows hf*8+j
    const float bn0 = bias[ 0 + lo];
    const float bn1 = bias[16 + lo];
    const float bn2 = bias[32 + lo];
    const float bn3 = bias[48 + lo];
    float* tb = &sTile[wv * 16 * TSTRIDE];
#pragma unroll
    for (int j = 0; j < 8; ++j) {
        const int M = hf * 8 + j;
        tb[M * TSTRIDE +  0 + lo] = a0[j] + bn0;
        tb[M * TSTRIDE + 16 + lo] = a1[j] + bn1;
        tb[M * TSTRIDE + 32 + lo] = a2[j] + bn2;
        tb[M * TSTRIDE + 48 + lo] = a3[j] + bn3;
    }
    __syncthreads();  // uniform: every wave has exactly one tile

    // softmax + KL: 2 lanes per row (lo = row, hf = class half), xor-16 combine
    const float* lrow = &tb[lo * TSTRIDE + hf * 32];
    float m = -INFINITY;
    for (int i = 0; i < 32; ++i) m = fmaxf(m, lrow[i]);
    m = fmaxf(m, __shfl_xor(m, 16, 32));

    float s = 0.f;
    for (int i = 0; i < 32; ++i) s += expf(lrow[i] - m);
    s += __shfl_xor(s, 16, 32);
    const float lse = logf(s);

    const float* prow = dist + (long)(row0 + lo) * NC + hf * 32;
    float kacc = 0.f;
    for (int i = 0; i < 32; ++i) {
        const float lq = fmaxf(lrow[i] - m - lse, LOG_EPS);   // log(clip(softmax,eps))
        const float p  = fmaxf(prow[i], EPSF);                // clip(target,eps)
        kacc += p * (logf(p) - lq);
    }
    kacc += __shfl_xor(kacc, 16, 32);
    if (hf == 0) neg_kl[row0 + lo] = -kacc;
}

// -----------------------------------------------------------------------------
// Kernel 2a/2b: two-stage global max of neg_kl (131072 -> 256 -> 1)
// -----------------------------------------------------------------------------
__global__ __launch_bounds__(256)
void rda_max_partial(const float* __restrict__ v, float* __restrict__ part)
{
    __shared__ float sm[256];
    const int tid  = threadIdx.x;
    const int base = blockIdx.x * 512;
    sm[tid] = fmaxf(v[base + tid], v[base + 256 + tid]);
    __syncthreads();
    for (int s = 128; s > 0; s >>= 1) {
        if (tid < s) sm[tid] = fmaxf(sm[tid], sm[tid + s]);
        __syncthreads();
    }
    if (tid == 0) part[blockIdx.x] = sm[0];
}

__global__ __launch_bounds__(256)
void rda_max_final(const float* __restrict__ part, float* __restrict__ gmax)
{
    __shared__ float sm[256];
    const int tid = threadIdx.x;
    sm[tid] = part[tid];
    __syncthreads();
    for (int s = 128; s > 0; s >>= 1) {
        if (tid < s) sm[tid] = fmaxf(sm[tid], sm[tid + s]);
        __syncthreads();
    }
    if (tid == 0) gmax[0] = sm[0];
}

// -----------------------------------------------------------------------------
// Kernel 3: per-b scores -> attended_weights (masked / UNmasked sum, faithful)
//           then weighted_inputs[b,h] = sum_t inputs[b,t,h] * w[b,t]
// out layout: [0 : B*H) weighted_inputs, [B*H : B*H + B*T) attended_weights
// -----------------------------------------------------------------------------
__global__ __launch_bounds__(256)
void rda_atten(const float* __restrict__ neg_kl,
               const float* __restrict__ gmax,
               const int*   __restrict__ lengths,
               const float* __restrict__ x,       // inputs [B][T][H]
               float* __restrict__ out)
{
    __shared__ float sw[TT];
    __shared__ float red[256];
    const int b = blockIdx.x;
    const int t = threadIdx.x;

    const float sc = expf(neg_kl[b * TT + t] - gmax[0]);
    red[t] = sc;
    __syncthreads();
    for (int s = 128; s > 0; s >>= 1) {
        if (t < s) red[t] += red[t + s];
        __syncthreads();
    }
    const float S   = red[0];                 // UNmasked sum (faithful to source)
    const int   len = lengths[b];
    const float aw  = ((t < len) ? sc : 0.f) / S;
    sw[t] = aw;
    out[(long)BB * HH + b * TT + t] = aw;     // attended_weights
    __syncthreads();

    // h = t : coalesced streaming pass over inputs[b]
    const float* xb = x + (long)b * TT * HH;
    float acc = 0.f;
#pragma unroll 4
    for (int tt = 0; tt < TT; ++tt) acc += xb[(long)tt * HH + t] * sw[tt];
    out[b * HH + t] = acc;
}

// -----------------------------------------------------------------------------
extern "C" void kernel_launch(void* const* d_in, const int* in_sizes, int n_in,
                              void* d_out, int out_size, void* d_ws, size_t ws_size,
                              hipStream_t stream)
{
    const float* x    = (const float*)d_in[0];  // inputs        [B,T,H]
    const float* xu   = (const float*)d_in[1];  // inputs_user   [B*T,DU]
    const int*   len  = (const int*)  d_in[2];  // input_lengths [B]
    const float* dist = (const float*)d_in[3];  // input_dist    [B*T,NC]
    const float* W    = (const float*)d_in[4];  // [KTOT,NC]
    const float* bias = (const float*)d_in[5];  // [NC]
    float*       out  = (float*)d_out;

    float* ws      = (float*)d_ws;
    float* neg_kl  = ws;                 // [ROWS]
    float* partmax = ws + ROWS;          // [256]
    float* gmax    = ws + ROWS + 256;    // [1]

    rda_gemm_kl   <<<ROWS / (16 * WAVES), 256, 0, stream>>>(x, xu, dist, W, bias, neg_kl);
    rda_max_partial<<<ROWS / 512,          256, 0, stream>>>(neg_kl, partmax);
    rda_max_final <<<1,                    256, 0, stream>>>(partmax, gmax);
    rda_atten     <<<BB,                   256, 0, stream>>>(neg_kl, gmax, len, x, out);
}